// GCN_80333068304388
// MI455X (gfx1250) — compile-verified
//
#include <hip/hip_runtime.h>
#include <math.h>

typedef float v2f __attribute__((ext_vector_type(2)));
typedef float v8f __attribute__((ext_vector_type(8)));

// ---------------------------------------------------------------- utilities
__global__ void k_fill(float* __restrict__ p, float v, int n) {
    int i = blockIdx.x * blockDim.x + threadIdx.x;
    if (i < n) p[i] = v;
}

__global__ void k_deg_accum(const int* __restrict__ dst, float* __restrict__ deg, int E) {
    int e = blockIdx.x * blockDim.x + threadIdx.x;
    if (e < E) unsafeAtomicAdd(&deg[dst[e]], 1.0f);
}

__global__ void k_rsqrt_inplace(float* __restrict__ p, int n) {
    int i = blockIdx.x * blockDim.x + threadIdx.x;
    if (i < n) p[i] = rsqrtf(p[i]);   // deg >= 1 always (self-loops)
}

__global__ void k_norm(const int* __restrict__ src, const int* __restrict__ dst,
                       const float* __restrict__ dis, float* __restrict__ norm, int E) {
    int e = blockIdx.x * blockDim.x + threadIdx.x;
    if (e < E) norm[e] = dis[src[e]] * dis[dst[e]];
}

// -------------------------------------------------- WMMA tall-skinny GEMM
// O[nrows, NCOL] = A[nrows, K] @ B[K, NCOL]   (fp32, V_WMMA_F32_16X16X4_F32)
// One wave per 16-row tile. Fragment layouts per CDNA5 ISA 7.12.2:
//  A 16x4 f32: lane L -> M = L&15, K offset = k0 + 2*(L>=16), 2 VGPRs (K,K+1)
//  B 4x16 f32: lane L -> N = L&15, K offset mirrored
//  C/D 16x16 : c[v] -> row v + 8*(L>=16), col L&15
// Tail handling: A row m only feeds D row m, so out-of-range lanes load a
// CLAMPED row (no divergent branch around the loads) and the per-row store
// guard drops the garbage. Keeps EXEC untouched through the WMMA chain.
template <int K, int NCOL>
__global__ __launch_bounds__(256) void k_gemm_wmma(const float* __restrict__ A,
                                                   const float* __restrict__ B,
                                                   float* __restrict__ O, int nrows) {
    int wave = threadIdx.x >> 5;
    int lane = threadIdx.x & 31;
    int tile = blockIdx.x * (blockDim.x >> 5) + wave;
    int base = tile << 4;
    if (base >= nrows) return;                 // uniform per wave
    int hi = lane >> 4;                        // 0 or 1
    int lo = lane & 15;                        // M for A-frag, N for B/C/D-frag

    int arow = base + lo;
    arow = arow < nrows ? arow : (nrows - 1);  // clamp: branch-free A loads
    const float* __restrict__ ap = A + (long)arow * K + 2 * hi;
    const float* __restrict__ bp = B + lo;
    bool bvalid = (lo < NCOL);

    v8f c = {0.f, 0.f, 0.f, 0.f, 0.f, 0.f, 0.f, 0.f};
#pragma unroll
    for (int k0 = 0; k0 < K; k0 += 4) {
        v2f a = *(const v2f*)(ap + k0);        // 8B-aligned, unconditional
        int ka = k0 + 2 * hi;
        v2f b;
        b.x = bvalid ? bp[(ka + 0) * NCOL] : 0.f;
        b.y = bvalid ? bp[(ka + 1) * NCOL] : 0.f;
        c = __builtin_amdgcn_wmma_f32_16x16x4_f32(false, a, false, b, (short)0, c,
                                                  false, false);
    }
#pragma unroll
    for (int v = 0; v < 8; ++v) {
        int row = base + v + 8 * hi;
        if (row < nrows && lo < NCOL) O[(long)row * NCOL + lo] = c[v];
    }
}

// ------------------------------------------------------------ edge scatter
// F threads per edge (F = 16 or 2): coalesced feature gather + fp32 atomics.
template <int F>
__global__ void k_scatter(const int* __restrict__ src, const int* __restrict__ dst,
                          const float* __restrict__ norm, const float* __restrict__ h,
                          float* __restrict__ acc, int E) {
    long gid = (long)blockIdx.x * blockDim.x + threadIdx.x;
    long e = gid / F;                 // F is power of two -> shift
    int f = (int)(gid - e * F);
    if (e < (long)E) {
        int s = src[e], d = dst[e];
        float v = norm[e] * h[(long)s * F + f];
        unsafeAtomicAdd(&acc[(long)d * F + f], v);
    }
}

// ------------------------------------------------------------- epilogues
// out = relu(acc + dis^2 * hx + b)   (F = 16)
__global__ void k_relu_ep(const float* __restrict__ acc, const float* __restrict__ hx,
                          const float* __restrict__ dis, const float* __restrict__ b,
                          float* __restrict__ out, int n) {
    int gid = blockIdx.x * blockDim.x + threadIdx.x;
    if (gid < n * 16) {
        int i = gid >> 4, f = gid & 15;
        float di = dis[i];
        float v = acc[gid] + di * di * hx[gid] + b[f];
        out[gid] = v > 0.f ? v : 0.f;
    }
}

// log_softmax over 2 classes
__global__ void k_lsm_ep(const float* __restrict__ acc, const float* __restrict__ hx,
                         const float* __restrict__ dis, const float* __restrict__ b,
                         float* __restrict__ out, int n) {
    int i = blockIdx.x * blockDim.x + threadIdx.x;
    if (i < n) {
        float d2 = dis[i] * dis[i];
        float z0 = acc[2 * i + 0] + d2 * hx[2 * i + 0] + b[0];
        float z1 = acc[2 * i + 1] + d2 * hx[2 * i + 1] + b[1];
        float m = fmaxf(z0, z1);
        float lse = m + __logf(__expf(z0 - m) + __expf(z1 - m));
        out[2 * i + 0] = z0 - lse;
        out[2 * i + 1] = z1 - lse;
    }
}

// ---------------------------------------------------------------- driver
extern "C" void kernel_launch(void* const* d_in, const int* in_sizes, int n_in,
                              void* d_out, int out_size, void* d_ws, size_t ws_size,
                              hipStream_t stream) {
    const float* x  = (const float*)d_in[0];
    const int*   ei = (const int*)d_in[1];
    const float* W1 = (const float*)d_in[2];
    const float* b1 = (const float*)d_in[3];
    const float* W2 = (const float*)d_in[4];
    const float* b2 = (const float*)d_in[5];
    const float* W3 = (const float*)d_in[6];
    const float* b3 = (const float*)d_in[7];

    const int N = in_sizes[0] / 128;
    const int E = in_sizes[1] / 2;
    const int* src = ei;
    const int* dst = ei + E;

    // workspace carve-out (256B-aligned regions)
    char* ws = (char*)d_ws;
    size_t off = 0;
    auto carve = [&](size_t bytes) -> void* {
        void* p = ws + off;
        off += (bytes + 255) & ~(size_t)255;
        return p;
    };
    float* dis  = (float*)carve((size_t)N * 4);        // deg -> dis (in place)
    float* norm = (float*)carve((size_t)E * 4);
    float* hx   = (float*)carve((size_t)N * 16 * 4);   // GEMM output (layer3 uses N*2)
    float* acc  = (float*)carve((size_t)N * 16 * 4);   // scatter accumulator
    float* h    = (float*)carve((size_t)N * 16 * 4);   // activations h1, then h2
    (void)ws_size; (void)n_in; (void)out_size;

    const int T = 256;
    auto blk = [](long n, int t) { return (unsigned)((n + t - 1) / t); };

    // normalization coefficients (computed once, reused by all 3 layers)
    k_fill<<<blk(N, T), T, 0, stream>>>(dis, 1.0f, N);                  // self-loop
    k_deg_accum<<<blk(E, T), T, 0, stream>>>(dst, dis, E);
    k_rsqrt_inplace<<<blk(N, T), T, 0, stream>>>(dis, N);
    k_norm<<<blk(E, T), T, 0, stream>>>(src, dst, dis, norm, E);

    const int ntile = (N + 15) / 16;
    const unsigned gblk = (unsigned)((ntile + 7) / 8);   // 8 waves / 256-thread block

    // ---- layer 1: [N,128] @ [128,16]
    k_gemm_wmma<128, 16><<<gblk, 256, 0, stream>>>(x, W1, hx, N);
    k_fill<<<blk((long)N * 16, T), T, 0, stream>>>(acc, 0.f, N * 16);
    k_scatter<16><<<blk((long)E * 16, T), T, 0, stream>>>(src, dst, norm, hx, acc, E);
    k_relu_ep<<<blk((long)N * 16, T), T, 0, stream>>>(acc, hx, dis, b1, h, N);

    // ---- layer 2: [N,16] @ [16,16]
    k_gemm_wmma<16, 16><<<gblk, 256, 0, stream>>>(h, W2, hx, N);
    k_fill<<<blk((long)N * 16, T), T, 0, stream>>>(acc, 0.f, N * 16);
    k_scatter<16><<<blk((long)E * 16, T), T, 0, stream>>>(src, dst, norm, hx, acc, E);
    k_relu_ep<<<blk((long)N * 16, T), T, 0, stream>>>(acc, hx, dis, b2, h, N);

    // ---- layer 3: [N,16] @ [16,2] + log_softmax
    k_gemm_wmma<16, 2><<<gblk, 256, 0, stream>>>(h, W3, hx, N);
    k_fill<<<blk((long)N * 2, T), T, 0, stream>>>(acc, 0.f, N * 2);
    k_scatter<2><<<blk((long)E * 2, T), T, 0, stream>>>(src, dst, norm, hx, acc, E);
    k_lsm_ep<<<blk(N, T), T, 0, stream>>>(acc, hx, dis, b3, (float*)d_out, N);
}